// GAT_74191265071505
// MI455X (gfx1250) — compile-verified
//
#include <hip/hip_runtime.h>
#include <hip/hip_bf16.h>

#define NEG_INF (-__builtin_inff())

typedef float v2f __attribute__((ext_vector_type(2)));
typedef float v8f __attribute__((ext_vector_type(8)));

// ---------------------------------------------------------------------------
// Init: zero accumulators, -inf the segment-max buffers.
// ---------------------------------------------------------------------------
__global__ void k_init(float* __restrict__ m1, float* __restrict__ dn1,
                       float* __restrict__ acc1, float* __restrict__ m2,
                       float* __restrict__ dn2, float* __restrict__ acc2, int N) {
    long long t = (long long)blockIdx.x * blockDim.x + threadIdx.x;
    long long n64 = (long long)N * 64;
    if (t < n64)      acc1[t] = 0.f;
    if (t < 4LL * N) { m1[t] = NEG_INF; dn1[t] = 0.f; }
    if (t < 2LL * N)  acc2[t] = 0.f;
    if (t < N)       { m2[t] = NEG_INF; dn2[t] = 0.f; }
}

// ---------------------------------------------------------------------------
// Layer-1 GEMM: h1[N,64] = x[N,128] @ W1[128,64], full-precision f32 WMMA.
// One wave per 16x16 output tile, 4 waves per block (covers the 64 cols).
// A 16x4 f32 layout: lanes 0-15 hold K={k0,k0+1}, lanes 16-31 K={k0+2,k0+3}.
// ---------------------------------------------------------------------------
__global__ __launch_bounds__(128)
void k_gemm1_wmma(const float* __restrict__ x, const float* __restrict__ W1,
                  float* __restrict__ h1, int N) {
    const int wave = threadIdx.x >> 5;   // 0..3 -> 16-col tile
    const int lane = threadIdx.x & 31;
    const int half = lane >> 4;          // 0 or 1
    const int l    = lane & 15;
    const int row0 = blockIdx.x * 16;
    const int col  = wave * 16 + l;      // B/D column for this lane
    int row = row0 + l;
    if (row >= N) row = N - 1;           // clamp reads for ragged last tile
    const float* xrow = x + (size_t)row * 128;

    v8f acc = {};
    for (int k0 = 0; k0 < 128; k0 += 4) {
        const int k = k0 + 2 * half;
        v2f a, b;
        a.x = xrow[k + 0];
        a.y = xrow[k + 1];
        b.x = W1[(size_t)(k + 0) * 64 + col];
        b.y = W1[(size_t)(k + 1) * 64 + col];
        acc = __builtin_amdgcn_wmma_f32_16x16x4_f32(
            /*neg_a=*/false, a, /*neg_b=*/false, b,
            /*c_mod=*/(short)0, acc, /*reuse_a=*/false, /*reuse_b=*/false);
    }
    // D layout: VGPR v holds M = v (lanes 0-15) / v+8 (lanes 16-31), N = l.
    // Wave-uniform bound check: full tiles take the unguarded path (always,
    // when N % 16 == 0), so no per-element exec-mask churn in the hot path.
    float* dp = h1 + (size_t)(row0 + 8 * half) * 64 + col;
    if (row0 + 16 <= N) {
#pragma unroll
        for (int v = 0; v < 8; ++v) dp[(size_t)v * 64] = acc[v];
    } else {
#pragma unroll
        for (int v = 0; v < 8; ++v) {
            if (row0 + v + 8 * half < N) dp[(size_t)v * 64] = acc[v];
        }
    }
}

// ---------------------------------------------------------------------------
// Per-node attention scores for layer 1: s[n,h] = sum_c h1[n,h,c]*a[h,c]
// ---------------------------------------------------------------------------
__global__ void k_scores1(const float* __restrict__ h1, const float* __restrict__ as,
                          const float* __restrict__ ad, float* __restrict__ ss,
                          float* __restrict__ sd, int N) {
    int t = blockIdx.x * blockDim.x + threadIdx.x;
    if (t >= N * 4) return;
    int n = t >> 2, h = t & 3;
    const float4* hp = (const float4*)(h1 + (size_t)n * 64 + h * 16);
    const float4* ap = (const float4*)(as + h * 16);
    const float4* bp = (const float4*)(ad + h * 16);
    float a = 0.f, b = 0.f;
#pragma unroll
    for (int i = 0; i < 4; ++i) {
        float4 hv = hp[i], av = ap[i], bv = bp[i];
        a += hv.x * av.x + hv.y * av.y + hv.z * av.z + hv.w * av.w;
        b += hv.x * bv.x + hv.y * bv.y + hv.z * bv.z + hv.w * bv.w;
    }
    ss[t] = a; sd[t] = b;
}

// Float atomic max via int/uint ordering trick (sign-correct, incl. +/-0).
__device__ __forceinline__ void atomicMaxF(float* addr, float v) {
    unsigned u = __float_as_uint(v);
    if (u & 0x80000000u) atomicMin((unsigned*)addr, u);
    else                 atomicMax((int*)addr, __float_as_int(v));
}

__device__ __forceinline__ float lrelu(float s) { return s > 0.f ? s : 0.2f * s; }

// ---------------------------------------------------------------------------
// Layer-1 edge pass A: segment max of leaky_relu(s_src+s_dst) over dst.
// One thread per (edge, head). Edges e>=E are self-loops (src=dst=e-E).
// ---------------------------------------------------------------------------
__global__ void k_edge_max1(const int* __restrict__ ei, const float* __restrict__ ss,
                            const float* __restrict__ sd, float* __restrict__ m1,
                            int E, int N) {
    long long t = (long long)blockIdx.x * blockDim.x + threadIdx.x;
    long long ET4 = 4LL * ((long long)E + N);
    if (t >= ET4) return;
    int h = (int)(t & 3);
    long long e = t >> 2;
    int src, dst;
    if (e < E) { src = ei[e]; dst = ei[(size_t)E + e]; } else { src = dst = (int)(e - E); }
    float s = lrelu(ss[(size_t)src * 4 + h] + sd[(size_t)dst * 4 + h]);
    atomicMaxF(&m1[(size_t)dst * 4 + h], s);
}

// ---------------------------------------------------------------------------
// Layer-1 edge pass B: fused denom += exp(e-m) and acc[dst] += h1[src]*exp(e-m).
// One thread per (edge, head), 16 channels via float4 gathers (b128 loads);
// normalization deferred to the node epilogue. All atomics are L2-resident
// (whole working set ~30 MB << 192 MB L2).
// ---------------------------------------------------------------------------
__global__ void k_edge_acc1(const int* __restrict__ ei, const float* __restrict__ ss,
                            const float* __restrict__ sd, const float* __restrict__ m1,
                            const float* __restrict__ h1, float* __restrict__ dn1,
                            float* __restrict__ acc1, int E, int N) {
    long long t = (long long)blockIdx.x * blockDim.x + threadIdx.x;
    long long ET4 = 4LL * ((long long)E + N);
    if (t >= ET4) return;
    int h = (int)(t & 3);
    long long e = t >> 2;
    int src, dst;
    if (e < E) { src = ei[e]; dst = ei[(size_t)E + e]; } else { src = dst = (int)(e - E); }
    float s  = lrelu(ss[(size_t)src * 4 + h] + sd[(size_t)dst * 4 + h]);
    float ex = expf(s - m1[(size_t)dst * 4 + h]);
    atomicAdd(&dn1[(size_t)dst * 4 + h], ex);
    const float4* hs = (const float4*)(h1 + (size_t)src * 64 + h * 16);
    float* ad = acc1 + (size_t)dst * 64 + h * 16;
#pragma unroll
    for (int i = 0; i < 4; ++i) {
        float4 v = hs[i];
        atomicAdd(&ad[4 * i + 0], v.x * ex);
        atomicAdd(&ad[4 * i + 1], v.y * ex);
        atomicAdd(&ad[4 * i + 2], v.z * ex);
        atomicAdd(&ad[4 * i + 3], v.w * ex);
    }
}

// ---------------------------------------------------------------------------
// Layer-1 node epilogue: normalize, +bias, ELU. In-place on acc1 (-> h_elu).
// ---------------------------------------------------------------------------
__global__ void k_node1(float* __restrict__ acc1, const float* __restrict__ dn1,
                        const float* __restrict__ b1, int N) {
    long long t = (long long)blockIdx.x * blockDim.x + threadIdx.x;
    if (t >= (long long)N * 64) return;
    int n = (int)(t >> 6), c = (int)(t & 63), h = c >> 4;
    float v = acc1[t] / dn1[(size_t)n * 4 + h] + b1[c];
    acc1[t] = v > 0.f ? v : expm1f(v);
}

// ---------------------------------------------------------------------------
// Layer-2 projection (64x2 GEMM too skinny for WMMA) + attention scores.
// ---------------------------------------------------------------------------
__global__ void k_proj2(const float* __restrict__ helu, const float* __restrict__ W2,
                        const float* __restrict__ as2, const float* __restrict__ ad2,
                        float* __restrict__ h2, float* __restrict__ ss2,
                        float* __restrict__ sd2, int N) {
    int n = blockIdx.x * blockDim.x + threadIdx.x;
    if (n >= N) return;
    const float* hp = helu + (size_t)n * 64;
    float o0 = 0.f, o1 = 0.f;
#pragma unroll
    for (int c = 0; c < 64; ++c) { float v = hp[c]; o0 += v * W2[c * 2]; o1 += v * W2[c * 2 + 1]; }
    h2[(size_t)n * 2]     = o0;
    h2[(size_t)n * 2 + 1] = o1;
    ss2[n] = o0 * as2[0] + o1 * as2[1];
    sd2[n] = o0 * ad2[0] + o1 * ad2[1];
}

__global__ void k_edge_max2(const int* __restrict__ ei, const float* __restrict__ ss2,
                            const float* __restrict__ sd2, float* __restrict__ m2,
                            int E, int N) {
    long long e = (long long)blockIdx.x * blockDim.x + threadIdx.x;
    if (e >= (long long)E + N) return;
    int src, dst;
    if (e < E) { src = ei[e]; dst = ei[(size_t)E + e]; } else { src = dst = (int)(e - E); }
    atomicMaxF(&m2[dst], lrelu(ss2[src] + sd2[dst]));
}

__global__ void k_edge_acc2(const int* __restrict__ ei, const float* __restrict__ ss2,
                            const float* __restrict__ sd2, const float* __restrict__ m2,
                            const float* __restrict__ h2, float* __restrict__ dn2,
                            float* __restrict__ acc2, int E, int N) {
    long long e = (long long)blockIdx.x * blockDim.x + threadIdx.x;
    if (e >= (long long)E + N) return;
    int src, dst;
    if (e < E) { src = ei[e]; dst = ei[(size_t)E + e]; } else { src = dst = (int)(e - E); }
    float ex = expf(lrelu(ss2[src] + sd2[dst]) - m2[dst]);
    atomicAdd(&dn2[dst], ex);
    atomicAdd(&acc2[(size_t)dst * 2],     h2[(size_t)src * 2]     * ex);
    atomicAdd(&acc2[(size_t)dst * 2 + 1], h2[(size_t)src * 2 + 1] * ex);
}

// ---------------------------------------------------------------------------
// Final epilogue: normalize, +b2, log_softmax over the 2 classes.
// ---------------------------------------------------------------------------
__global__ void k_out(const float* __restrict__ acc2, const float* __restrict__ dn2,
                      const float* __restrict__ b2, float* __restrict__ out, int N) {
    int n = blockIdx.x * blockDim.x + threadIdx.x;
    if (n >= N) return;
    float d  = dn2[n];
    float v0 = acc2[(size_t)n * 2]     / d + b2[0];
    float v1 = acc2[(size_t)n * 2 + 1] / d + b2[1];
    float m  = fmaxf(v0, v1);
    float lse = m + logf(expf(v0 - m) + expf(v1 - m));
    out[(size_t)n * 2]     = v0 - lse;
    out[(size_t)n * 2 + 1] = v1 - lse;
}

static inline unsigned nblk(long long n, int b) { return (unsigned)((n + b - 1) / b); }

extern "C" void kernel_launch(void* const* d_in, const int* in_sizes, int n_in,
                              void* d_out, int out_size, void* d_ws, size_t ws_size,
                              hipStream_t stream) {
    const float* x     = (const float*)d_in[0];
    const int*   ei    = (const int*)  d_in[1];
    const float* W1    = (const float*)d_in[2];
    const float* asrc1 = (const float*)d_in[3];
    const float* adst1 = (const float*)d_in[4];
    const float* b1    = (const float*)d_in[5];
    const float* W2    = (const float*)d_in[6];
    const float* asrc2 = (const float*)d_in[7];
    const float* adst2 = (const float*)d_in[8];
    const float* b2    = (const float*)d_in[9];
    float* out = (float*)d_out;

    const int N = in_sizes[0] / 128;   // F_IN = 128
    const int E = in_sizes[1] / 2;     // edge_index is [2,E]
    const long long ET = (long long)E + N;

    // Workspace layout (floats): everything fits in L2 (~30 MB total).
    float* ws   = (float*)d_ws;
    size_t nl   = (size_t)N;
    float* h1   = ws;                 // [N,64]
    float* ss1  = h1  + nl * 64;      // [N,4]
    float* sd1  = ss1 + nl * 4;       // [N,4]
    float* m1   = sd1 + nl * 4;       // [N,4]
    float* dn1  = m1  + nl * 4;       // [N,4]
    float* acc1 = dn1 + nl * 4;       // [N,64] -> becomes h_elu in place
    float* h2   = acc1 + nl * 64;     // [N,2]
    float* ss2  = h2  + nl * 2;       // [N]
    float* sd2  = ss2 + nl;           // [N]
    float* m2   = sd2 + nl;           // [N]
    float* dn2  = m2  + nl;           // [N]
    float* acc2 = dn2 + nl;           // [N,2]

    k_init<<<nblk((long long)N * 64, 256), 256, 0, stream>>>(m1, dn1, acc1, m2, dn2, acc2, N);
    k_gemm1_wmma<<<(N + 15) / 16, 128, 0, stream>>>(x, W1, h1, N);
    k_scores1<<<nblk(4LL * N, 256), 256, 0, stream>>>(h1, asrc1, adst1, ss1, sd1, N);
    k_edge_max1<<<nblk(4 * ET, 256), 256, 0, stream>>>(ei, ss1, sd1, m1, E, N);
    k_edge_acc1<<<nblk(4 * ET, 256), 256, 0, stream>>>(ei, ss1, sd1, m1, h1, dn1, acc1, E, N);
    k_node1<<<nblk((long long)N * 64, 256), 256, 0, stream>>>(acc1, dn1, b1, N);
    k_proj2<<<nblk(N, 256), 256, 0, stream>>>(acc1, W2, asrc2, adst2, h2, ss2, sd2, N);
    k_edge_max2<<<nblk(ET, 256), 256, 0, stream>>>(ei, ss2, sd2, m2, E, N);
    k_edge_acc2<<<nblk(ET, 256), 256, 0, stream>>>(ei, ss2, sd2, m2, h2, dn2, acc2, E, N);
    k_out<<<nblk(N, 256), 256, 0, stream>>>(acc2, dn2, b2, out, N);
}